// DSRLoss_57930518889022
// MI455X (gfx1250) — compile-verified
//
#include <hip/hip_runtime.h>
#include <hip/hip_bf16.h>
#include <math.h>

// DSR loss: R = rowsum(W .* X) over [262144 x 128] f32, then a parallelized
// affine-EMA scan (eta=0.01) producing D_t, output = -mean(D).
//
// Phase 1 is HBM-bound (256 MB -> ~11us @ 23.3 TB/s). Row reduction done with
// V_WMMA_F32_16X16X4_F32 (full f32 precision, CDNA5 matrix pipe). Since the
// B-matrix is all-ones, the mapping of asset indices onto (WMMA#, K-slot) is
// arbitrary: lanes 0-15 own assets [0,64) of their row, lanes 16-31 own
// [64,128), letting every lane stream its half-row with 16B float4 loads
// (one b128 load feeds two WMMA fragments -> half the vmem instructions).
//
// Phase 2 uses chunked affine-scan decomposition: the serial dependency is an
// affine map with constant decay, so 1024 chunks of 256 steps run in parallel,
// one tiny serial pass chains the chunk transforms, then chunks replay the
// exact reference recurrence and emit deterministic partial sums.

#define ETA   0.01f
#define EPS   1e-8f
#define BATCH   262144
#define NASSET  128
#define NCHUNK  1024
#define CHUNK   256     // NCHUNK * CHUNK == BATCH

typedef __attribute__((ext_vector_type(2))) float v2f;
typedef __attribute__((ext_vector_type(8))) float v8f;

// ---------------------------------------------------------------------------
// Phase 1: per-row dot product via WMMA f32 16x16x4.
// One wave32 handles 16 consecutive rows (time steps). B = all ones => every
// column of D is the row sum. C/D layout: lane 0 holds M=0..7 in c[0..7];
// lane 16 holds M=8..15.
// ---------------------------------------------------------------------------
__global__ __launch_bounds__(256) void dsr_rowdot_wmma(
    const float* __restrict__ W, const float* __restrict__ X,
    float* __restrict__ R) {
  const int lane = threadIdx.x & 31;
  const int waveId = (blockIdx.x * blockDim.x + threadIdx.x) >> 5;
  const size_t t0 = (size_t)waveId * 16;           // 16 rows per wave
  const int m    = lane & 15;                      // row within tile
  const int half = lane >> 4;                      // 0: assets [0,64), 1: [64,128)
  const size_t base = (t0 + m) * NASSET + (size_t)half * 64;  // 16B aligned
  const float4* wrow = (const float4*)(W + base);
  const float4* xrow = (const float4*)(X + base);

  v8f c0 = {0.f, 0.f, 0.f, 0.f, 0.f, 0.f, 0.f, 0.f};
  v8f c1 = {0.f, 0.f, 0.f, 0.f, 0.f, 0.f, 0.f, 0.f};
  v2f bones; bones.x = 1.0f; bones.y = 1.0f;

  // 64 assets per lane-half = 16 float4 loads per matrix; each load feeds two
  // K=4 WMMAs. Two accumulators break the D->C dependency chain.
#pragma unroll
  for (int i = 0; i < 16; i += 2) {
    float4 w0 = wrow[i];
    float4 x0 = xrow[i];
    float4 w1 = wrow[i + 1];
    float4 x1 = xrow[i + 1];
    v2f a00; a00.x = w0.x * x0.x; a00.y = w0.y * x0.y;
    v2f a01; a01.x = w0.z * x0.z; a01.y = w0.w * x0.w;
    v2f a10; a10.x = w1.x * x1.x; a10.y = w1.y * x1.y;
    v2f a11; a11.x = w1.z * x1.z; a11.y = w1.w * x1.w;
    c0 = __builtin_amdgcn_wmma_f32_16x16x4_f32(false, a00, false, bones,
                                               (short)0, c0, false, false);
    c1 = __builtin_amdgcn_wmma_f32_16x16x4_f32(false, a01, false, bones,
                                               (short)0, c1, false, false);
    c0 = __builtin_amdgcn_wmma_f32_16x16x4_f32(false, a10, false, bones,
                                               (short)0, c0, false, false);
    c1 = __builtin_amdgcn_wmma_f32_16x16x4_f32(false, a11, false, bones,
                                               (short)0, c1, false, false);
  }

  if (lane == 0) {
#pragma unroll
    for (int g = 0; g < 8; ++g) R[t0 + g] = c0[g] + c1[g];
  } else if (lane == 16) {
#pragma unroll
    for (int g = 0; g < 8; ++g) R[t0 + 8 + g] = c0[g] + c1[g];
  }
}

// ---------------------------------------------------------------------------
// Phase 2a: each chunk's local forced response (affine scan, start = 0).
// ---------------------------------------------------------------------------
__global__ void dsr_chunk_local(const float* __restrict__ R,
                                float* __restrict__ tA,
                                float* __restrict__ tB) {
  const int j = blockIdx.x * blockDim.x + threadIdx.x;
  if (j >= NCHUNK) return;
  const float* r = R + (size_t)j * CHUNK;
  float a = 0.f, b = 0.f;
#pragma unroll 4
  for (int i = 0; i < CHUNK; ++i) {
    const float rv = r[i];
    a += ETA * (rv - a);
    b += ETA * (rv * rv - b);
  }
  tA[j] = a;
  tB[j] = b;
}

// ---------------------------------------------------------------------------
// Phase 2b: chain the 1024 chunk transforms: state' = c^CHUNK * state + t_j.
// Single thread, 1024 serial steps (microseconds), fully deterministic.
// ---------------------------------------------------------------------------
__global__ void dsr_chunk_carry(const float* __restrict__ tA,
                                const float* __restrict__ tB,
                                const float* __restrict__ A0p,
                                const float* __restrict__ B0p,
                                float* __restrict__ Astart,
                                float* __restrict__ Bstart) {
  if (threadIdx.x != 0 || blockIdx.x != 0) return;
  const float f = __powf(1.0f - ETA, (float)CHUNK);
  float a = A0p[0], b = B0p[0];
  for (int j = 0; j < NCHUNK; ++j) {
    Astart[j] = a;
    Bstart[j] = b;
    a = f * a + tA[j];
    b = f * b + tB[j];
  }
}

// ---------------------------------------------------------------------------
// Phase 2c: replay exact reference recurrence per chunk, emit partial sum(D).
// ---------------------------------------------------------------------------
__global__ void dsr_chunk_D(const float* __restrict__ R,
                            const float* __restrict__ Astart,
                            const float* __restrict__ Bstart,
                            float* __restrict__ partial) {
  const int j = blockIdx.x * blockDim.x + threadIdx.x;
  if (j >= NCHUNK) return;
  const float* r = R + (size_t)j * CHUNK;
  float a = Astart[j], b = Bstart[j], s = 0.f;
#pragma unroll 4
  for (int i = 0; i < CHUNK; ++i) {
    const float rv  = r[i];
    const float dA  = ETA * (rv - a);
    const float dB  = ETA * (rv * rv - b);
    const float var = fmaxf(b - a * a, EPS);
    const float D   = (b * dA - 0.5f * a * dB) / (var * sqrtf(var));
    s += D;
    a += dA;
    b += dB;
  }
  partial[j] = s;
}

// ---------------------------------------------------------------------------
// Final: fixed-order reduction (deterministic; no float atomics).
// ---------------------------------------------------------------------------
__global__ void dsr_finalize(const float* __restrict__ partial,
                             float* __restrict__ out) {
  if (threadIdx.x != 0 || blockIdx.x != 0) return;
  float s = 0.f;
  for (int j = 0; j < NCHUNK; ++j) s += partial[j];
  out[0] = -s / (float)BATCH;
}

extern "C" void kernel_launch(void* const* d_in, const int* in_sizes, int n_in,
                              void* d_out, int out_size, void* d_ws, size_t ws_size,
                              hipStream_t stream) {
  (void)in_sizes; (void)n_in; (void)out_size; (void)ws_size;

  const float* W   = (const float*)d_in[0];   // weights      [BATCH*NASSET]
  const float* X   = (const float*)d_in[1];   // next_returns [BATCH*NASSET]
  const float* A0p = (const float*)d_in[2];   // scalar
  const float* B0p = (const float*)d_in[3];   // scalar
  float* out = (float*)d_out;

  // Workspace layout (floats): R[BATCH] | tA[NCHUNK] | tB | Astart | Bstart | partial
  float* R       = (float*)d_ws;
  float* tA      = R + BATCH;
  float* tB      = tA + NCHUNK;
  float* Astart  = tB + NCHUNK;
  float* Bstart  = Astart + NCHUNK;
  float* partial = Bstart + NCHUNK;

  // Phase 1: 262144 rows / 16 rows per wave = 16384 waves; 8 waves per block.
  const int waves = BATCH / 16;
  const int blk1  = 256;
  dsr_rowdot_wmma<<<waves * 32 / blk1, blk1, 0, stream>>>(W, X, R);

  // Phase 2a: 1024 chunk threads.
  dsr_chunk_local<<<NCHUNK / 256, 256, 0, stream>>>(R, tA, tB);

  // Phase 2b: serial chunk-carry chain (1 wave).
  dsr_chunk_carry<<<1, 32, 0, stream>>>(tA, tB, A0p, B0p, Astart, Bstart);

  // Phase 2c: replay chunks with true start states.
  dsr_chunk_D<<<NCHUNK / 256, 256, 0, stream>>>(R, Astart, Bstart, partial);

  // Final deterministic reduction.
  dsr_finalize<<<1, 32, 0, stream>>>(partial, out);
}